// _ResGCN_4252017623277
// MI455X (gfx1250) — compile-verified
//
#include <hip/hip_runtime.h>

#define N_NODES 100000
#define N_EDGES 800000
#define N_GRAPHS 64
#define HID 256
#define ALPHA 0.2f

typedef __attribute__((ext_vector_type(2))) float v2f;
typedef __attribute__((ext_vector_type(8))) float v8f;

__device__ __forceinline__ float lrelu(float v) { return v > 0.f ? v : ALPHA * v; }

// ---------------------------------------------------------------------------
// Tiled WMMA GEMM: out[M,256] = epilogue(A[M,KDIM] @ W[KDIM,256] + bias)
//   GATHER: A row index comes from gather[] (embedding lookup fusion)
//   ACT:    leaky-relu epilogue
//   RES:    out = res + result (residual add)
// Block = 256 threads = 8 waves; each wave computes one 16x16 tile with
// v_wmma_f32_16x16x4_f32. A tile staged in LDS, shared by all 8 waves.
// ---------------------------------------------------------------------------
template <int KDIM, bool GATHER, bool ACT, bool RES>
__global__ __launch_bounds__(256) void gemm_wmma_k(
    const float* __restrict__ A, const float* __restrict__ W,
    const float* __restrict__ bias, const int* __restrict__ gather,
    const float* __restrict__ res, float* __restrict__ out) {
  __shared__ float sA[16 * KDIM];
  const int tid = threadIdx.x;
  const int tileM = blockIdx.x;

  // cooperative stage of A tile (16 rows x KDIM) into LDS
  for (int i = tid; i < 16 * KDIM; i += 256) {
    const int r = i / KDIM;
    const int c = i - r * KDIM;
    const int row = tileM * 16 + r;
    const int srow = GATHER ? gather[row] : row;
    sA[i] = A[(size_t)srow * KDIM + c];
  }
  __syncthreads();

  const int wave = tid >> 5;
  const int lane = tid & 31;
  const int half = lane >> 4;   // K-half selector per ISA A/B layout
  const int l16  = lane & 15;
  const int colT = blockIdx.y * 8 + wave;
  const int colbase = colT * 16;

  v8f c = {};
#pragma unroll 4
  for (int kk = 0; kk < KDIM; kk += 4) {
    v2f a, b;
    // A 16x4 f32 layout: lanes 0-15 -> K=kk,kk+1 ; lanes 16-31 -> K=kk+2,kk+3
    a.x = sA[l16 * KDIM + kk + 2 * half];
    a.y = sA[l16 * KDIM + kk + 2 * half + 1];
    // B 4x16 f32 layout (transpose-symmetric): row K, col = colbase+l16
    b.x = W[(size_t)(kk + 2 * half) * HID + colbase + l16];
    b.y = W[(size_t)(kk + 2 * half + 1) * HID + colbase + l16];
    c = __builtin_amdgcn_wmma_f32_16x16x4_f32(false, a, false, b, (short)0, c,
                                              false, false);
  }

  const int col = colbase + l16;
  const float bv = bias[col];
#pragma unroll
  for (int r = 0; r < 8; ++r) {
    // C/D layout: VGPR r -> M = r (lanes 0-15) or r+8 (lanes 16-31)
    const int row = tileM * 16 + r + 8 * half;
    float v = c[r] + bv;
    if (ACT) v = lrelu(v);
    const size_t idx = (size_t)row * HID + col;
    if (RES) v += res[idx];
    out[idx] = v;
  }
}

// ---------------------------------------------------------------------------
// elementwise / edge kernels
// ---------------------------------------------------------------------------
__global__ void zero_k(float* p, int n) {
  int i = blockIdx.x * 256 + threadIdx.x;
  if (i < n) p[i] = 0.f;
}

__global__ void deg_k(const int* __restrict__ src, float* deg) {
  int e = blockIdx.x * 256 + threadIdx.x;
  if (e < N_EDGES) atomicAdd(&deg[src[e]], 1.f);
}

__global__ void scale_k(const float* __restrict__ deg, float* scale) {
  int i = blockIdx.x * 256 + threadIdx.x;
  if (i < N_NODES) scale[i] = 1.f / (deg[i] + 1.f);
}

// y = lrelu(x); ys = y*scale; acc1 = acc2 = 0   (grid = N*H/256 exact)
__global__ void prepare_k(const float* __restrict__ x,
                          const float* __restrict__ scale, float* y, float* ys,
                          float* acc1, float* acc2) {
  size_t i = (size_t)blockIdx.x * 256 + threadIdx.x;
  float v = lrelu(x[i]);
  y[i] = v;
  ys[i] = v * scale[i >> 8];
  acc1[i] = 0.f;
  acc2[i] = 0.f;
}

// acc1[dst] += ys[src];  acc2[src] += y[dst]   (64 threads x float4 per edge)
__global__ void scatter_k(const float* __restrict__ ys,
                          const float* __restrict__ y,
                          const int* __restrict__ src,
                          const int* __restrict__ dst, float* acc1,
                          float* acc2) {
  long long id = (long long)blockIdx.x * 256 + threadIdx.x;
  int e = (int)(id >> 6);
  int f = (int)(id & 63) * 4;
  int s = src[e], d = dst[e];
  const float4 v1 = *reinterpret_cast<const float4*>(ys + (size_t)s * HID + f);
  const float4 v2 = *reinterpret_cast<const float4*>(y + (size_t)d * HID + f);
  float* p1 = acc1 + (size_t)d * HID + f;
  float* p2 = acc2 + (size_t)s * HID + f;
  atomicAdd(p1 + 0, v1.x); atomicAdd(p1 + 1, v1.y);
  atomicAdd(p1 + 2, v1.z); atomicAdd(p1 + 3, v1.w);
  atomicAdd(p2 + 0, v2.x); atomicAdd(p2 + 1, v2.y);
  atomicAdd(p2 + 2, v2.z); atomicAdd(p2 + 3, v2.w);
}

// m = (acc1 + ys) + scale*(acc2 + y) + y, written in-place into acc1
__global__ void combine_k(const float* __restrict__ ys,
                          const float* __restrict__ y,
                          const float* __restrict__ scale,
                          const float* __restrict__ acc2, float* acc1m) {
  size_t i = (size_t)blockIdx.x * 256 + threadIdx.x;
  float m = acc1m[i] + ys[i] + scale[i >> 8] * (acc2[i] + y[i]) + y[i];
  acc1m[i] = m;
}

// pooled[graph_ids[node]] += lrelu(x[node])
__global__ void pool_k(const float* __restrict__ x, const int* __restrict__ gid,
                       float* pooled) {
  size_t i = (size_t)blockIdx.x * 256 + threadIdx.x;
  int node = (int)(i >> 8);
  int f = (int)(i & 255);
  atomicAdd(&pooled[(size_t)gid[node] * HID + f], lrelu(x[i]));
}

// out[g] = lrelu(pooled[g] @ W1 + b1) @ W2 + b2   (one block per graph)
__global__ __launch_bounds__(256) void post_k(const float* __restrict__ pooled,
                                              const float* __restrict__ W1,
                                              const float* __restrict__ b1,
                                              const float* __restrict__ W2,
                                              const float* __restrict__ b2,
                                              float* out) {
  __shared__ float sp[HID];
  __shared__ float red[HID];
  const int g = blockIdx.x, j = threadIdx.x;
  sp[j] = pooled[(size_t)g * HID + j];
  __syncthreads();
  float s = 0.f;
#pragma unroll 8
  for (int k = 0; k < HID; ++k) s = fmaf(sp[k], W1[(size_t)k * HID + j], s);
  red[j] = lrelu(s + b1[j]) * W2[j];
  __syncthreads();
  for (int off = 128; off > 0; off >>= 1) {
    if (j < off) red[j] += red[j + off];
    __syncthreads();
  }
  if (j == 0) out[g] = red[0] + b2[0];
}

// ---------------------------------------------------------------------------
extern "C" void kernel_launch(void* const* d_in, const int* in_sizes, int n_in,
                              void* d_out, int out_size, void* d_ws,
                              size_t ws_size, hipStream_t stream) {
  (void)in_sizes; (void)n_in; (void)out_size; (void)ws_size;
  const int* op_ids = (const int*)d_in[0];
  const int* esrc = (const int*)d_in[1];
  const int* edst = (const int*)d_in[2];
  const int* gid = (const int*)d_in[3];
  const float* emb = (const float*)d_in[4];
  const float* pW1 = (const float*)d_in[5];
  const float* pb1 = (const float*)d_in[6];
  const float* pW2 = (const float*)d_in[7];
  const float* pb2 = (const float*)d_in[8];
  const float* gW1 = (const float*)d_in[9];
  const float* gb1 = (const float*)d_in[10];
  const float* gW2 = (const float*)d_in[11];
  const float* gb2 = (const float*)d_in[12];
  const float* qW1 = (const float*)d_in[13];
  const float* qb1 = (const float*)d_in[14];
  const float* qW2 = (const float*)d_in[15];
  const float* qb2 = (const float*)d_in[16];
  float* out = (float*)d_out;

  const size_t NH = (size_t)N_NODES * HID;
  float* ws = (float*)d_ws;
  float* x = ws;
  float* y = ws + NH;
  float* ys = ws + 2 * NH;      // also doubles as GEMM intermediate
  float* acc1 = ws + 3 * NH;    // becomes m in place
  float* acc2 = ws + 4 * NH;
  float* scale = ws + 5 * NH;
  float* deg = scale + N_NODES;
  float* pooled = deg + N_NODES;

  // ---- degree / normalization ----
  zero_k<<<(N_NODES + 255) / 256, 256, 0, stream>>>(deg, N_NODES);
  deg_k<<<(N_EDGES + 255) / 256, 256, 0, stream>>>(esrc, deg);
  scale_k<<<(N_NODES + 255) / 256, 256, 0, stream>>>(deg, scale);

  const dim3 gg(N_NODES / 16, 2);  // 6250 x 2, 8 waves/block -> 16 col tiles
  // ---- prenet: x = lrelu(emb[op_ids] @ W1 + b1) @ W2 + b2 ----
  gemm_wmma_k<64, true, true, false>
      <<<gg, 256, 0, stream>>>(emb, pW1, pb1, op_ids, nullptr, ys);
  gemm_wmma_k<HID, false, false, false>
      <<<gg, 256, 0, stream>>>(ys, pW2, pb2, nullptr, nullptr, x);

  const int NHB = (int)(NH / 256);          // 100000 blocks, exact
  const int EB = (N_EDGES * 64) / 256;      // 200000 blocks, exact
  for (int i = 0; i < 3; ++i) {
    prepare_k<<<NHB, 256, 0, stream>>>(x, scale, y, ys, acc1, acc2);
    scatter_k<<<EB, 256, 0, stream>>>(ys, y, esrc, edst, acc1, acc2);
    combine_k<<<NHB, 256, 0, stream>>>(ys, y, scale, acc2, acc1);
    gemm_wmma_k<HID, false, true, false><<<gg, 256, 0, stream>>>(
        acc1, gW1 + (size_t)i * HID * HID, gb1 + i * HID, nullptr, nullptr, ys);
    gemm_wmma_k<HID, false, false, true><<<gg, 256, 0, stream>>>(
        ys, gW2 + (size_t)i * HID * HID, gb2 + i * HID, nullptr, x, x);
  }

  // ---- pool + head ----
  zero_k<<<(N_GRAPHS * HID + 255) / 256, 256, 0, stream>>>(pooled,
                                                           N_GRAPHS * HID);
  pool_k<<<NHB, 256, 0, stream>>>(x, gid, pooled);
  post_k<<<N_GRAPHS, 256, 0, stream>>>(pooled, qW1, qb1, qW2, qb2, out);
}